// CrossEntropyLoss_18081812316881
// MI455X (gfx1250) — compile-verified
//
#include <hip/hip_runtime.h>
#include <hip/hip_bf16.h>

// CE loss: B=8, C=3, H=W=1024. Pure streaming reduction (160MB in -> 4B out),
// HBM-bound (~7us floor at 23.3 TB/s). Stage 1: vectorized grid-stride
// partial sums (one float per block). Stage 2: single wave32 folds the 2048
// partials with chained v_wmma_f32_16x16x4_f32 (B = ones => row-sum reducer).

typedef __attribute__((ext_vector_type(2))) float v2f;
typedef __attribute__((ext_vector_type(8))) float v8f;

#define HWSZ   (1024 * 1024)          // H*W
#define NPIX   (8 * HWSZ)             // B*H*W
#define NGRP   (NPIX / 4)             // float4 groups
#define BLOCKS 2048
#define TPB    256
#define CLAMP_MIN 1e-8f

__device__ __forceinline__ float pix_ce(float a0, float a1, float a2, int c,
                                        float mk, float w0, float w1, float w2) {
    float m  = fmaxf(a0, fmaxf(a1, a2));
    float e0 = __expf(a0 - m);
    float e1 = __expf(a1 - m);
    float e2 = __expf(a2 - m);
    float s  = e0 + e1 + e2;
    float ey = (c == 0) ? e0 : ((c == 1) ? e1 : e2);
    float wy = (c == 0) ? w0 : ((c == 1) ? w1 : w2);
    float p  = ey / s;
    float lp = __logf(fmaxf(p, CLAMP_MIN));
    return wy * lp * mk;   // sign folded into final scale
}

__global__ void __launch_bounds__(TPB)
ce_partial_kernel(const float* __restrict__ x, const int* __restrict__ y,
                  const float* __restrict__ weight, const float* __restrict__ mask,
                  float* __restrict__ partials) {
    const float w0 = weight[0], w1 = weight[1], w2 = weight[2];
    const int tid    = blockIdx.x * TPB + threadIdx.x;
    const int stride = BLOCKS * TPB;

    float acc = 0.0f;
    for (int g = tid; g < NGRP; g += stride) {
        const int p  = g << 2;            // pixel index, multiple of 4
        const int b  = p >> 20;           // p / HWSZ
        const int hw = p & (HWSZ - 1);
        const int xb = b * (3 * HWSZ) + hw;

        const float4* px0 = (const float4*)(x + xb);
        const float4  x0  = px0[0];
        const float4  x1  = *(const float4*)(x + xb + HWSZ);
        const float4  x2  = *(const float4*)(x + xb + 2 * HWSZ);
        const int4    yv  = *(const int4*)(y + p);
        const float4  mv  = *(const float4*)(mask + p);

        // gfx1250 vmem prefetch path (speculative; OOB silently dropped)
        __builtin_prefetch((const void*)(px0 + stride), 0, 0);

        acc += pix_ce(x0.x, x1.x, x2.x, yv.x, mv.x, w0, w1, w2);
        acc += pix_ce(x0.y, x1.y, x2.y, yv.y, mv.y, w0, w1, w2);
        acc += pix_ce(x0.z, x1.z, x2.z, yv.z, mv.z, w0, w1, w2);
        acc += pix_ce(x0.w, x1.w, x2.w, yv.w, mv.w, w0, w1, w2);
    }

    // wave32 reduction
    #pragma unroll
    for (int off = 16; off > 0; off >>= 1)
        acc += __shfl_xor(acc, off, 32);

    __shared__ float smem[TPB / 32];
    const int lane = threadIdx.x & 31;
    const int wid  = threadIdx.x >> 5;
    if (lane == 0) smem[wid] = acc;
    __syncthreads();

    if (wid == 0) {
        float v = (lane < (TPB / 32)) ? smem[lane] : 0.0f;
        #pragma unroll
        for (int off = 16; off > 0; off >>= 1)
            v += __shfl_xor(v, off, 32);
        if (lane == 0) partials[blockIdx.x] = v;
    }
}

// Single wave (32 threads). Fold BLOCKS partials with chained WMMA:
// D = A x ones(4x16) + C  => each D row holds a 16-wide-replicated chunk row
// sum, so sum(D) == 16 * sum(partials). Scale = -1/(16*N).
__global__ void __launch_bounds__(32)
ce_final_kernel(const float* __restrict__ partials, float* __restrict__ out,
                float scale) {
    const int lane = threadIdx.x;
    v8f c = {};
    v2f bones;
    bones[0] = 1.0f;
    bones[1] = 1.0f;

    #pragma unroll 4
    for (int ch = 0; ch < BLOCKS / 64; ++ch) {
        const float2 d = *(const float2*)(partials + ch * 64 + lane * 2);
        v2f a;
        a[0] = d.x;
        a[1] = d.y;
        // 8 args: (neg_a, A, neg_b, B, c_mod, C, reuse_a, reuse_b)
        c = __builtin_amdgcn_wmma_f32_16x16x4_f32(
                false, a, false, bones, (short)0, c, false, false);
    }

    float s = 0.0f;
    #pragma unroll
    for (int i = 0; i < 8; ++i) s += c[i];
    #pragma unroll
    for (int off = 16; off > 0; off >>= 1)
        s += __shfl_xor(s, off, 32);

    if (lane == 0) out[0] = s * scale;
}

extern "C" void kernel_launch(void* const* d_in, const int* in_sizes, int n_in,
                              void* d_out, int out_size, void* d_ws, size_t ws_size,
                              hipStream_t stream) {
    const float* x      = (const float*)d_in[0];
    const int*   y      = (const int*)d_in[1];
    const float* weight = (const float*)d_in[2];
    const float* mask   = (const float*)d_in[3];
    float*       out    = (float*)d_out;
    float*       parts  = (float*)d_ws;   // BLOCKS floats = 8KB scratch

    ce_partial_kernel<<<BLOCKS, TPB, 0, stream>>>(x, y, weight, mask, parts);

    const float scale = -1.0f / (16.0f * (float)NPIX);
    ce_final_kernel<<<1, 32, 0, stream>>>(parts, out, scale);
}